// CompressDCT_4801773436989
// MI455X (gfx1250) — compile-verified
//
#include <hip/hip_runtime.h>

typedef __attribute__((ext_vector_type(2))) float v2f;
typedef __attribute__((ext_vector_type(8))) float v8f;

#define LSTRIDE 20   // floats; 80B rows: 16B aligned, conflict-free both orientations
#define WIMG    112
#define TROW    7    // tile columns per image row, processed by one wave

// 8-point DCT-II matrix entries (0.5*cos(k*pi*(2m+1)/16), row0 = sqrt(1/8))
#define DA 0.49039264020161522f
#define DB 0.46193976625564337f
#define DC 0.41573480615127262f
#define DDv 0.35355339059327376f
#define DE 0.27778511650980111f
#define DF 0.19134171618254489f
#define DG 0.09754516100806414f

__device__ __constant__ float DCT8[64] = {
  DDv, DDv, DDv, DDv, DDv, DDv, DDv, DDv,
  DA,  DC,  DE,  DG, -DG, -DE, -DC, -DA,
  DB,  DF, -DF, -DB, -DB, -DF,  DF,  DB,
  DC, -DG, -DA, -DE,  DE,  DA,  DG, -DC,
  DDv,-DDv,-DDv, DDv, DDv,-DDv,-DDv, DDv,
  DE, -DA,  DG,  DC, -DC, -DG,  DA, -DE,
  DF, -DB,  DB, -DF, -DF,  DB, -DB,  DF,
  DG, -DE,  DC, -DA,  DA, -DC,  DE, -DG
};

// Dbig = diag(D, D): zero when (r,c) in different 8-halves
__device__ __forceinline__ float dbig(int r, int c) {
  float v = DCT8[((r & 7) << 3) | (c & 7)];
  return ((r ^ c) & 8) ? 0.0f : v;
}

// Full 16x16 (K=16) f32 matmul: 4 chained V_WMMA_F32_16X16X4_F32
__device__ __forceinline__ v8f wmma16(const v2f a[4], const v2f b[4]) {
  v8f acc = {0.f, 0.f, 0.f, 0.f, 0.f, 0.f, 0.f, 0.f};
#pragma unroll
  for (int kk = 0; kk < 4; ++kk)
    acc = __builtin_amdgcn_wmma_f32_16x16x4_f32(
        false, a[kk], false, b[kk], (short)0, acc, false, false);
  return acc;
}

// Store D-layout accumulator TRANSPOSED into LDS: element (m+8h, col) -> lds[col][m+8h].
// Contiguous per lane -> two b128 stores.
__device__ __forceinline__ void store_T(float* lds, v8f c, int col, int half) {
  int base = col * LSTRIDE + 8 * half;
  float4 lo = {c[0], c[1], c[2], c[3]};
  float4 hi = {c[4], c[5], c[6], c[7]};
  *reinterpret_cast<float4*>(lds + base)     = lo;
  *reinterpret_cast<float4*>(lds + base + 4) = hi;
}

// Load B fragments (4x16 chunks) of the row-major 16x16 tile in LDS.
// Wave32 B layout: lanes 0-15 hold K={4kk,4kk+1}, lanes 16-31 hold K={4kk+2,4kk+3}.
__device__ __forceinline__ void load_b(const float* lds, v2f b[4], int col, int half) {
#pragma unroll
  for (int kk = 0; kk < 4; ++kk) {
    int r0 = 4 * kk + 2 * half;
    b[kk].x = lds[r0 * LSTRIDE + col];
    b[kk].y = lds[(r0 + 1) * LSTRIDE + col];
  }
}

__global__ __launch_bounds__(32) void dct8x8_wmma_kernel(
    const float* __restrict__ x, const float* __restrict__ qt,
    float* __restrict__ out) {
  __shared__ float lds[16 * LSTRIDE];
  const int lane = threadIdx.x;
  const int half = lane >> 4;
  const int col  = lane & 15;

  // One tile-row (7 tiles of 16x16) per wave/block: streams contiguous image rows
  int th  = blockIdx.x % TROW;       // tile row within image
  int img = blockIdx.x / TROW;       // n*C + c
  long base0 = (long)img * (WIMG * WIMG) + (long)(th * 16) * WIMG;

  // ---- per-wave constants (amortized over 7 tiles) ----
  // Quant row for this lane's elements (X2^T layout: row=lane&7, col=vgpr m)
  float qv[8], rq[8];
#pragma unroll
  for (int m = 0; m < 8; ++m) {
    qv[m] = qt[((lane & 7) << 3) | m];
    rq[m] = __builtin_amdgcn_rcpf(qv[m]);  // v_rcp_f32, not full IEEE div
  }

  // Constant A fragments: Dbig and Dbig^T rows for M = col
  // A layout: lanes 0-15 hold K={4kk,4kk+1}, lanes 16-31 hold K={4kk+2,4kk+3}
  v2f aD[4], aDT[4];
#pragma unroll
  for (int kk = 0; kk < 4; ++kk) {
    int j0 = 4 * kk + 2 * half;
    aD[kk]  = (v2f){dbig(col, j0),  dbig(col, j0 + 1)};   // Dbig[M][j]
    aDT[kk] = (v2f){dbig(j0, col),  dbig(j0 + 1, col)};   // Dbig^T[M][j] = Dbig[j][M]
  }

  // Precompute loop-invariant lane addressing
  const int rS  = lane >> 1;          // output row handled by this lane
  const int cbS = (lane & 1) << 3;    // output col base (0 or 8)

  for (int tw = 0; tw < TROW; ++tw) {
    long base = base0 + tw * 16;
    const float* __restrict__ X = x + base;
    float* __restrict__ Y = out + base;

    // Load X directly into B fragments (rows = 64B contiguous per lane-half)
    v2f b[4];
#pragma unroll
    for (int kk = 0; kk < 4; ++kk) {
      int r0 = 4 * kk + 2 * half;
      b[kk].x = X[r0 * WIMG + col];
      b[kk].y = X[(r0 + 1) * WIMG + col];
    }

    v8f t1 = wmma16(aD, b);      // T1 = Dbig * X
    store_T(lds, t1, col, half); // LDS = T1^T
    __syncthreads();
    load_b(lds, b, col, half);   // B = T1^T
    __syncthreads();

    v8f w2 = wmma16(aD, b);      // Dbig * T1^T = (T1 * Dbig^T)^T = X2^T
    // Quantize: element = X2[col][m+8h] -> q = q_table[col%8][m]
#pragma unroll
    for (int m = 0; m < 8; ++m)
      w2[m] = rintf(w2[m] * rq[m]) * qv[m];  // v_rndne == jnp.round

    store_T(lds, w2, col, half); // LDS = (Xq^T)^T = Xq
    __syncthreads();
    load_b(lds, b, col, half);   // B = Xq
    __syncthreads();

    v8f t2 = wmma16(aDT, b);     // T2 = Dbig^T * Xq
    store_T(lds, t2, col, half); // LDS = T2^T
    __syncthreads();
    load_b(lds, b, col, half);   // B = T2^T
    __syncthreads();

    v8f w4 = wmma16(aDT, b);     // Dbig^T * T2^T = (T2 * Dbig)^T = Y^T
    store_T(lds, w4, col, half); // LDS = Y (row-major)
    __syncthreads();

    // Coalesced output: lane pair covers one 64B row
    const float4* s = reinterpret_cast<const float4*>(lds + rS * LSTRIDE + cbS);
    float4* d = reinterpret_cast<float4*>(Y + (long)rS * WIMG + cbS);
    d[0] = s[0];
    d[1] = s[1];

    __syncthreads();  // WAR: next iteration's ds_store vs this iteration's final reads
  }
}

extern "C" void kernel_launch(void* const* d_in, const int* in_sizes, int n_in,
                              void* d_out, int out_size, void* d_ws, size_t ws_size,
                              hipStream_t stream) {
  const float* x  = (const float*)d_in[0];
  const float* qt = (const float*)d_in[1];
  float* out = (float*)d_out;
  int images = in_sizes[0] / (WIMG * WIMG);  // N*C = 4096
  int blocks = images * TROW;                // one tile-row (7 tiles) per wave
  dct8x8_wmma_kernel<<<blocks, 32, 0, stream>>>(x, qt, out);
}